// CRF_26379689132766
// MI455X (gfx1250) — compile-verified
//
#include <hip/hip_runtime.h>
#include <hip/hip_bf16.h>
#include <stdint.h>

// Problem constants (from reference setup_inputs): B=256, T=512, N=128
#define B_ 256
#define T_ 512
#define N_ 128
#define CHUNK 32               // timesteps per TDM tile (32*128*4 = 16 KB)
#define NCHUNK (T_ / CHUNK)

typedef uint32_t u32x4 __attribute__((ext_vector_type(4)));
typedef int      i32x8 __attribute__((ext_vector_type(8)));
typedef int      i32x4 __attribute__((ext_vector_type(4)));

#if __has_builtin(__builtin_amdgcn_tensor_load_to_lds) && __has_builtin(__builtin_amdgcn_s_wait_tensorcnt)
#define HAVE_TDM 1
#else
#define HAVE_TDM 0
#endif

// Generic pointers into the LDS aperture carry the wave-relative LDS byte
// offset in their low 32 bits (ISA 10.2).
__device__ __forceinline__ uint32_t lds_offset_of(const void* p) {
  return (uint32_t)(uintptr_t)p;
}

#if HAVE_TDM
// TDM descriptor per CDNA5 ISA ch.8: 2D tile of `tile_rows` x N_ f32 elements,
// row stride N_, loaded Global -> LDS. Groups 2/3 present with harmless dims
// (tensor_dim2..4 = 1, tile_dim2..4 = 0). 6-arg builtin (clang-23 /
// therock-10.0): extra int32x8 arg passed zero-filled.
__device__ __forceinline__ void tdm_load_rows(const float* gsrc, uint32_t lds_off,
                                              uint32_t rows_avail, uint32_t tile_rows) {
  uint64_t ga = (uint64_t)(uintptr_t)gsrc;
  u32x4 g0;
  g0[0] = 1u;                                             // count=1, user mode
  g0[1] = lds_off;                                        // lds_addr [63:32]
  g0[2] = (uint32_t)ga;                                   // global_addr [95:64]
  g0[3] = ((uint32_t)(ga >> 32) & 0x01FFFFFFu) | (2u << 30); // addr[120:96] | type=2
  i32x8 g1;
  g1[0] = (int)(2u << 16);                                // data_size = 2 (4 bytes)
  g1[1] = (int)((uint32_t)N_ << 16);                      // tensor_dim0 = 128 (bits 63:48)
  g1[2] = (int)((rows_avail & 0xFFFFu) << 16);            // tensor_dim1 lo16 (bits 95:80)
  g1[3] = (int)((uint32_t)N_ << 16);                      // tile_dim0 = 128 (bits 127:112)
  g1[4] = (int)(tile_rows & 0xFFFFu);                     // tile_dim1, tile_dim2 = 0
  g1[5] = N_;                                             // tensor_dim0_stride = 128
  g1[6] = 0;
  g1[7] = 0;
  i32x4 g2 = {1, 1, 0, 0};                                // tensor_dim2 = 1, tensor_dim3 = 1
  i32x4 g3 = {0, (1 << 16), 0, 0};                        // tensor_dim4 = 1, tile_dim4 = 0
  i32x8 g4 = {0, 0, 0, 0, 0, 0, 0, 0};                    // extra group (zero-filled)
  __builtin_amdgcn_tensor_load_to_lds(g0, g1, g2, g3, g4, 0);
}
#endif

__global__ __launch_bounds__(N_) void crf_viterbi_kernel(
    const float* __restrict__ logits,      // [B, T, N] f32
    const float* __restrict__ trans,       // [N, N]    f32
    const int*   __restrict__ lens,        // [B]       i32
    int*         __restrict__ out)         // [B, T]    i32
{
  // Dynamic LDS layout (99,712 B total; fits 320 KB WGP LDS):
  //   xbuf : 2 * CHUNK * N floats  (32 KB)  double-buffered logit tiles (TDM)
  //   st0/1: 2 * N floats          ( 1 KB)  double-buffered Viterbi state
  //   bp   : (T-1) * N bytes       (64 KB)  backpointers (LDS-resident)
  //   tags : T bytes               (0.5 KB) backtrace result staging
  extern __shared__ char smem[];
  float*         xbuf = (float*)smem;
  float*         st0  = (float*)(smem + 2 * CHUNK * N_ * 4);
  float*         st1  = st0 + N_;
  unsigned char* bp   = (unsigned char*)(st1 + N_);
  unsigned char* tags = bp + (T_ - 1) * N_;

  const int  j     = threadIdx.x;          // tag owned by this lane (wave32 x 4)
  const int  b     = blockIdx.x;           // sequence
  const bool wave0 = (j >> 5) == 0;
  const float* Lb  = logits + (size_t)b * T_ * N_;
  const int  len   = lens[b];              // in [1, T]

  // Transition column j -> 128 VGPRs (read once; L2-resident across blocks).
  float tc[N_];
  #pragma unroll
  for (int i = 0; i < N_; ++i) tc[i] = trans[i * N_ + j];

  // ---- bring in chunk 0 ----
#if HAVE_TDM
  if (wave0) {
    tdm_load_rows(Lb, lds_offset_of(xbuf), T_, CHUNK);
    __builtin_amdgcn_s_wait_tensorcnt(0);
  }
#else
  {
    const float4* gs = (const float4*)Lb;
    float4* ds = (float4*)xbuf;
    #pragma unroll
    for (int k = 0; k < (CHUNK * N_ / 4) / N_; ++k) ds[j + N_ * k] = gs[j + N_ * k];
  }
#endif
  __syncthreads();

  float* stc = st0;
  float* stn = st1;
  stc[j] = xbuf[j];                        // state0 = logits[b, 0, :]
  __syncthreads();

  // ---- forward Viterbi over chunks of CHUNK timesteps ----
  for (int c = 0; c < NCHUNK; ++c) {
    const int t0 = c * CHUNK;
    if (t0 >= len) break;                  // uniform per block
#if HAVE_TDM
    if (wave0 && (t0 + CHUNK) < len && (c + 1) < NCHUNK) {
      tdm_load_rows(Lb + (size_t)(t0 + CHUNK) * N_,
                    lds_offset_of(xbuf + ((c + 1) & 1) * CHUNK * N_),
                    (uint32_t)(T_ - (t0 + CHUNK)), CHUNK);
    }
#endif
    const float* xb   = xbuf + (c & 1) * CHUNK * N_;
    const int tstart  = (c == 0) ? 1 : t0;
    const int tend    = (t0 + CHUNK < len) ? (t0 + CHUNK) : len;
    for (int t = tstart; t < tend; ++t) {
      // new[j] = max_i(state[i] + trans[i][j]).
      // 4 independent (best,arg) chains (one per float4 lane) -> ~4x shorter
      // dependency chain; merged with index-priority so the FIRST maximal i
      // wins overall (matches jnp.argmax). Within a chain, strict > keeps the
      // first occurrence of that chain's residue class.
      const float4* sp = (const float4*)stc;   // broadcast LDS reads
      float b0 = -3.4028235e38f, b1 = b0, b2 = b0, b3 = b0;
      int   a0 = 0, a1 = 1, a2 = 2, a3 = 3;
      #pragma unroll
      for (int k = 0; k < N_ / 4; ++k) {
        float4 s = sp[k];
        float v;
        v = s.x + tc[4 * k + 0]; if (v > b0) { b0 = v; a0 = 4 * k + 0; }
        v = s.y + tc[4 * k + 1]; if (v > b1) { b1 = v; a1 = 4 * k + 1; }
        v = s.z + tc[4 * k + 2]; if (v > b2) { b2 = v; a2 = 4 * k + 2; }
        v = s.w + tc[4 * k + 3]; if (v > b3) { b3 = v; a3 = 4 * k + 3; }
      }
      // merge (value desc, index asc) -- associative for first-max semantics
      if (b1 > b0 || (b1 == b0 && a1 < a0)) { b0 = b1; a0 = a1; }
      if (b3 > b2 || (b3 == b2 && a3 < a2)) { b2 = b3; a2 = a3; }
      if (b2 > b0 || (b2 == b0 && a2 < a0)) { b0 = b2; a0 = a2; }

      stn[j] = b0 + xb[(t - t0) * N_ + j];
      bp[(t - 1) * N_ + j] = (unsigned char)a0;
      __syncthreads();                     // one barrier/step (double-buffered state)
      float* tmp = stc; stc = stn; stn = tmp;
    }
#if HAVE_TDM
    if (wave0) __builtin_amdgcn_s_wait_tensorcnt(0);
    __syncthreads();
#else
    if ((t0 + CHUNK) < len && (c + 1) < NCHUNK) {
      const float4* gs = (const float4*)(Lb + (size_t)(t0 + CHUNK) * N_);
      float4* ds = (float4*)(xbuf + ((c + 1) & 1) * CHUNK * N_);
      #pragma unroll
      for (int k = 0; k < (CHUNK * N_ / 4) / N_; ++k) ds[j + N_ * k] = gs[j + N_ * k];
    }
    __syncthreads();
#endif
  }

  // ---- final argmax + serial backtrace (LDS-latency, not HBM) ----
  if (j == 0) {
    float bestv = stc[0];
    int   cur   = 0;
    for (int i = 1; i < N_; ++i) {
      float v = stc[i];
      if (v > bestv) { bestv = v; cur = i; }
    }
    tags[len - 1] = (unsigned char)cur;
    for (int s = len - 2; s >= 0; --s) {
      cur = bp[s * N_ + cur];              // bp slot s holds step t = s+1
      tags[s] = (unsigned char)cur;
    }
  }
  __syncthreads();

  // ---- masked int32 output ----
  int* ob = out + (size_t)b * T_;
  #pragma unroll
  for (int r = 0; r < T_ / N_; ++r) {
    int t = j + r * N_;
    ob[t] = (t < len) ? (int)tags[t] : 0;
  }
}

extern "C" void kernel_launch(void* const* d_in, const int* in_sizes, int n_in,
                              void* d_out, int out_size, void* d_ws, size_t ws_size,
                              hipStream_t stream) {
  const float* logits = (const float*)d_in[0];   // [256,512,128] f32
  const float* trans  = (const float*)d_in[1];   // [128,128]     f32
  const int*   lens   = (const int*)d_in[2];     // [256]         i32
  int* out = (int*)d_out;                        // [256,512]     i32 (bit-stored)

  const size_t smem = (size_t)(2 * CHUNK * N_ * 4)   // xbuf
                    + (size_t)(2 * N_ * 4)           // st0/st1
                    + (size_t)((T_ - 1) * N_)        // bp
                    + (size_t)T_;                    // tags
  crf_viterbi_kernel<<<B_, N_, smem, stream>>>(logits, trans, lens, out);
}